// CPC_Loss_30640296690406
// MI455X (gfx1250) — compile-verified
//
#include <hip/hip_runtime.h>

#define B_    128
#define T_    512
#define E_    128
#define K_    4
#define NNEG_ 64

// ce LDS row: 512 floats + TDM pad of 2 dwords every 256 dwords -> stride 516
// (516 % 64 == 4 -> the 16 rows hit distinct banks on column reads)
#define CE_RS 516
// neg/be LDS rows: 128 floats + TDM pad of 4 dwords every 128 -> stride 132
#define EP    132

#define LOG65 4.1743873f   // loss of a fully masked row: -log(1/(NNEG+1))

typedef __attribute__((ext_vector_type(2))) float        v2f;
typedef __attribute__((ext_vector_type(8))) float        v8f;
typedef __attribute__((ext_vector_type(4))) unsigned int v4u;
typedef __attribute__((ext_vector_type(8))) int          v8i;
typedef __attribute__((ext_vector_type(4))) int          v4i;

// ---------- DPP 16-lane reductions (VALU, co-executes with WMMA pipe) ----------
template <int CTRL>
__device__ __forceinline__ float dpp_mov(float v) {
    int x = __builtin_bit_cast(int, v);
    x = __builtin_amdgcn_update_dpp(x, x, CTRL, 0xF, 0xF, true);
    return __builtin_bit_cast(float, x);
}
__device__ __forceinline__ float red16_max(float v) {
    v = fmaxf(v, dpp_mov<0xB1>(v));    // quad_perm xor1
    v = fmaxf(v, dpp_mov<0x4E>(v));    // quad_perm xor2
    v = fmaxf(v, dpp_mov<0x141>(v));   // row_half_mirror (pairs of 8)
    v = fmaxf(v, dpp_mov<0x140>(v));   // row_mirror (pairs of 16)
    return v;
}
__device__ __forceinline__ float red16_sum(float v) {
    v += dpp_mov<0xB1>(v);
    v += dpp_mov<0x4E>(v);
    v += dpp_mov<0x141>(v);
    v += dpp_mov<0x140>(v);
    return v;
}

// ---------- LDS byte offset of a __shared__ object (generic -> AS3) ----------
__device__ __forceinline__ unsigned lds_off(const void* p) {
    return (unsigned)(unsigned long long)(const __attribute__((address_space(3))) void*)p;
}

// ---------- Tensor Data Mover: 2D tile load, D# per ISA 8.3/8.4 ----------
__device__ __forceinline__ void tdm_load_2d(unsigned lds_addr, unsigned long long ga,
                                            unsigned tensor_d0, unsigned tensor_d1,
                                            unsigned tile_d0, unsigned tile_d1,
                                            unsigned long long stride0,
                                            unsigned pad_interval, unsigned pad_amount) {
    v4u g0;
    g0[0] = 1u;                                                  // count=1, user D#
    g0[1] = lds_addr;                                            // LDS byte address
    g0[2] = (unsigned)ga;                                        // global addr lo
    g0[3] = (unsigned)((ga >> 32) & 0x01FFFFFFull) | 0x80000000u; // addr[56:32] | type=2
    v8i g1;
    g1[0] = (int)((2u << 16)                                     // data_size = 4B
                | (1u << 20)                                     // pad_enable
                | (pad_interval << 22) | (pad_amount << 25));
    g1[1] = (int)((tensor_d0 & 0xFFFFu) << 16);                  // [31:16]=td0.lo
    g1[2] = (int)((tensor_d0 >> 16) | ((tensor_d1 & 0xFFFFu) << 16));
    g1[3] = (int)((tensor_d1 >> 16) | (tile_d0 << 16));
    g1[4] = (int)tile_d1;                                        // tile_dim2 = 0
    g1[5] = (int)(unsigned)stride0;                              // dim0 stride lo
    g1[6] = (int)((unsigned)(stride0 >> 32) & 0xFFFFu);          // stride hi | stride1=0
    g1[7] = 0;
    v4i z4 = {};
    v8i z8 = {};
    __builtin_amdgcn_tensor_load_to_lds(g0, g1, z4, z4, z8, 0);
}

__global__ void cpc_zero_out(float* out) { out[0] = 0.0f; }

// neg[b][n][e] = base_payload.reshape(B*T, E)[sample_ids[b][n]][e]
__global__ void cpc_gather_neg(const float* __restrict__ base,
                               const int* __restrict__ sids,
                               float* __restrict__ neg) {
    int idx = blockIdx.x * blockDim.x + threadIdx.x;   // B*NNEG*(E/4) float4s
    int c   = idx & 31;
    int bn  = idx >> 5;
    int row = sids[bn];
    const float4* src = (const float4*)base;
    float4*       dst = (float4*)neg;
    dst[bn * (E_ / 4) + c] = src[row * (E_ / 4) + c];
}

__global__ void __launch_bounds__(128)
cpc_main(const float* __restrict__ base, const float* __restrict__ mcp,
         const int* __restrict__ seqlen, const float* __restrict__ neg,
         float* __restrict__ out) {
    // ce raw [s][e][k] layout (K fastest, as in memory); TDM inserts the pads.
    __shared__ __align__(16) float ceS[16 * CE_RS];        // ~33 KB
    __shared__ __align__(16) float negS[NNEG_][EP];        // ~33 KB
    __shared__ __align__(16) float beS[19][EP];            // ~10 KB
    __shared__ float posS[K_][16];
    __shared__ float redS[K_][16][4][2];

    const int tid  = threadIdx.x;
    const int lane = tid & 31;
    const int w    = tid >> 5;
    const int l16  = lane & 15;
    const int s0   = blockIdx.x * 16;
    const int b    = blockIdx.y;
    const int sl   = seqlen[b];

    // ---- stage everything with the Tensor Data Mover (one wave issues) ----
    if (w == 0) {
        // ce tile: 16 rows of 512 floats (E*K), pad 2 dwords every 256 -> stride 516
        tdm_load_2d(lds_off(ceS),
                    (unsigned long long)(const void*)(mcp + (size_t)(b * T_ + s0) * (E_ * K_)),
                    E_ * K_, (unsigned)(T_ - s0), E_ * K_, 16, E_ * K_,
                    /*interval 2^(7+1)=256 dw*/ 7, /*amount 1+1=2 dw*/ 1);
        // negatives: 64 rows of 128 floats, pad 4 dwords every 128 -> stride 132
        tdm_load_2d(lds_off(negS),
                    (unsigned long long)(const void*)(neg + (size_t)b * NNEG_ * E_),
                    E_, NNEG_, E_, NNEG_, E_,
                    /*interval 2^(6+1)=128 dw*/ 6, /*amount 3+1=4 dw*/ 3);
        // base rows t = s0+1 .. s0+19; tensor_dim1 clamp zero-fills rows past T
        tdm_load_2d(lds_off(beS),
                    (unsigned long long)(const void*)(base + (size_t)(b * T_ + s0 + 1) * E_),
                    E_, (unsigned)(T_ - (s0 + 1)), E_, 19, E_,
                    6, 3);
        __builtin_amdgcn_s_wait_tensorcnt(0);
    }
    __syncthreads();

    float lossAcc = 0.0f;
    // WMMA f32 16x16x4 fragment mapping (ISA 7.12.2): lane = M (A) / N (B) index,
    // VGPR0/1 hold K {ko, ko+1} with ko = (lane>=16)*2.
    const int row = l16;
    const int ko  = (lane >> 4) << 1;

    for (int i = 0; i < K_; ++i) {
        // ---- positive logits: each 16-lane half owns 2 rows ----
        {
            int h = lane >> 4;
#pragma unroll
            for (int jj = 0; jj < 2; ++jj) {
                int sr = w * 4 + h * 2 + jj;
                float p = 0.0f;
#pragma unroll
                for (int q = 0; q < 8; ++q) {
                    int e = l16 + q * 16;
                    p += ceS[sr * CE_RS + e * 4 + i + ((q >= 4) ? 2 : 0)] * beS[sr + i][e];
                }
                p = red16_sum(p);
                if (l16 == 0) posS[i][sr] = p;
            }
        }

        // ---- neg logits: 16(M=s) x 16(N=this wave's negs) x 128(K=e) ----
        const float* aR = &ceS[row * CE_RS + ko * 4 + i];   // ce[row][kb+ko][i]
        const float* bR = &negS[w * 16 + row][ko];
        v8f acc = {};
#pragma unroll
        for (int kb = 0; kb < E_; kb += 4) {
            const int pad = (kb >= 64) ? 2 : 0;             // mid-row TDM pad
            v2f a, bb;
            a.x  = aR[kb * 4 + pad];
            a.y  = aR[kb * 4 + pad + 4];
            bb.x = bR[kb];
            bb.y = bR[kb + 1];
            acc = __builtin_amdgcn_wmma_f32_16x16x4_f32(
                false, a, false, bb, (short)0, acc, false, false);
        }

        // ---- per-row max / sum-exp over this wave's 16 columns (DPP, no DS) ----
        float mx[8], se[8];
#pragma unroll
        for (int r = 0; r < 8; ++r) {
            float v = acc[r];
            float m = red16_max(v);
            float s = red16_sum(__expf(v - m));
            mx[r] = m; se[r] = s;
        }
        if (l16 == 0) {
            int rbase = (lane >> 4) * 8;
#pragma unroll
            for (int r = 0; r < 8; ++r) {
                redS[i][rbase + r][w][0] = mx[r];
                redS[i][rbase + r][w][1] = se[r];
            }
        }
        __syncthreads();

        // ---- combine 4 wave-partials + positive into log-softmax loss ----
        if (tid < 16) {
            int sr = tid;
            if (s0 + sr < T_ - (i + 1)) {           // positions s >= T-(i+1) don't count
                float loss;
                if (s0 + sr >= sl) {
                    loss = LOG65;                    // masked row: logits are all zero
                } else {
                    float pos = posS[i][sr];
                    float m = pos;
#pragma unroll
                    for (int w2 = 0; w2 < 4; ++w2) m = fmaxf(m, redS[i][sr][w2][0]);
                    float tot = __expf(pos - m);
#pragma unroll
                    for (int w2 = 0; w2 < 4; ++w2)
                        tot += redS[i][sr][w2][1] * __expf(redS[i][sr][w2][0] - m);
                    loss = __logf(tot) + m - pos;
                }
                lossAcc += loss * (1.0f / ((float)K_ * (float)B_ * (float)(T_ - (i + 1))));
            }
        }
    }

    if (w == 0) {                       // partials live in lanes 0..15 of wave 0
        lossAcc = red16_sum(lossAcc);
        if (lane == 0) atomicAdd(out, lossAcc);
    }
}

extern "C" void kernel_launch(void* const* d_in, const int* in_sizes, int n_in,
                              void* d_out, int out_size, void* d_ws, size_t ws_size,
                              hipStream_t stream) {
    const float* base   = (const float*)d_in[0];   // (B, T, E) f32
    const float* mcp    = (const float*)d_in[1];   // (B, T, E, K) f32
    const int*   seqlen = (const int*)d_in[2];     // (B,) i32
    const int*   sids   = (const int*)d_in[3];     // (B, NNEG) i32
    float* out = (float*)d_out;                    // scalar f32
    float* neg = (float*)d_ws;                     // (B, NNEG, E) f32 = 4 MB

    cpc_zero_out<<<1, 1, 0, stream>>>(out);

    int gthreads = B_ * NNEG_ * (E_ / 4);          // 262144
    cpc_gather_neg<<<gthreads / 256, 256, 0, stream>>>(base, sids, neg);

    dim3 grid(T_ / 16, B_);
    cpc_main<<<grid, 128, 0, stream>>>(base, mcp, seqlen, neg, out);
}